// graphSAGE_15977278341798
// MI455X (gfx1250) — compile-verified
//
#include <hip/hip_runtime.h>

typedef float v2f __attribute__((ext_vector_type(2)));
typedef float v8f __attribute__((ext_vector_type(8)));

#define D 128

// ---------------------------------------------------------------------------
// Kernel 1: zero the accumulation workspace (agg[N*128] and deg[N]).
// Must run every call: atomics accumulate into it.
// ---------------------------------------------------------------------------
__global__ void sage_zero_ws(float* __restrict__ agg, float* __restrict__ deg,
                             int n_nodes) {
    int i = blockIdx.x * blockDim.x + threadIdx.x;
    int total = n_nodes * D;
    if (i < total)  agg[i] = 0.0f;
    if (i < n_nodes) deg[i] = 0.0f;
}

// ---------------------------------------------------------------------------
// Kernel 2: edge scatter. One wave (32 lanes) per edge.
// Lane l reads float4 of feat[src] (512B row fully coalesced, L2-resident)
// and atomically adds 4 floats into agg[dst]. Lane 0 counts the degree.
// feat (25.6MB) and agg (25.6MB) both fit in the 192MB L2, so this phase is
// L2/atomic-throughput bound, not HBM bound.
// ---------------------------------------------------------------------------
__global__ void __launch_bounds__(256)
sage_scatter(const float* __restrict__ feat,
             const int*   __restrict__ src,
             const int*   __restrict__ dst,
             float*       __restrict__ agg,
             float*       __restrict__ deg,
             int n_edges) {
    int tid  = blockIdx.x * blockDim.x + threadIdx.x;
    int e    = tid >> 5;
    int lane = tid & 31;
    if (e >= n_edges) return;
    int s = src[e];
    int d = dst[e];
    const float4 v = *(const float4*)(feat + (size_t)s * D + lane * 4);
    float* ab = agg + (size_t)d * D + lane * 4;
    atomicAdd(ab + 0, v.x);
    atomicAdd(ab + 1, v.y);
    atomicAdd(ab + 2, v.z);
    atomicAdd(ab + 3, v.w);
    if (lane == 0) atomicAdd(deg + d, 1.0f);
}

// ---------------------------------------------------------------------------
// Kernel 3: fused dual-GEMM with exact f32 WMMA (V_WMMA_F32_16X16X4_F32).
//   out = feat @ W_self^T + (agg * inv_deg) @ W_neigh^T + (b_self + b_neigh)
// One wave per 16-node row tile; 8 N-tiles (128 out channels) per wave.
// Both weight matrices staged in LDS (2 x 64KB / block). launch_bounds(256,1)
// relaxes the occupancy target (LDS already caps us at 2 blocks/WGP) so all
// 16 B fragments per K-chunk live in distinct VGPRs -> batched ds loads,
// staggered waits, back-to-back WMMAs. A loads are software-pipelined one
// chunk ahead.
//
// Layouts per CDNA5 ISA:
//   A 16x4 f32 : lane l -> M = l%16, VGPRs {K=2*(l/16), 2*(l/16)+1}
//   B 4x16 f32 : lane l -> N = l%16, VGPRs {K=2*(l/16), 2*(l/16)+1}
//   C/D 16x16  : vgpr v at lane l -> (M = v + 8*(l/16), N = l%16)
// ---------------------------------------------------------------------------
__global__ void __launch_bounds__(256, 1)
sage_wmma_gemm(const float* __restrict__ feat,
               const float* __restrict__ agg,
               const float* __restrict__ deg,
               const float* __restrict__ Wself,
               const float* __restrict__ bself,
               const float* __restrict__ Wneigh,
               const float* __restrict__ bneigh,
               float*       __restrict__ out,
               int n_nodes, int n_tiles) {
    extern __shared__ float smem[];          // [2 * 128 * 128] floats = 128KB

    // ---- cooperative weight staging (all 256 threads, then barrier) ----
    {
        const int tid = threadIdx.x;
        float4*       s4  = (float4*)smem;
        const float4* ws4 = (const float4*)Wself;
        const float4* wn4 = (const float4*)Wneigh;
        const int nvec = (D * D) / 4;        // 4096 float4 per matrix
#pragma unroll 4
        for (int i = tid; i < nvec; i += 256) {
            s4[i]        = ws4[i];
            s4[nvec + i] = wn4[i];
        }
    }
    __syncthreads();

    const int lane = threadIdx.x & 31;
    const int wave = threadIdx.x >> 5;
    const int tile = blockIdx.x * 8 + wave;   // wave-uniform

    if (tile < n_tiles) {                     // wave-uniform guard: EXEC all-1s inside
        const int nsub  = lane & 15;          // M for A loads, N for B loads / stores
        const int khalf = (lane >> 4) * 2;    // K offset within a K=4 chunk

        int mrow = tile * 16 + nsub;
        if (mrow >= n_nodes) mrow = n_nodes - 1;   // safe clamp (50000 % 16 == 0)

        // Per-lane neighbor-mean scale (lane's A rows all share node mrow).
        const float dg  = deg[mrow];
        const float inv = 1.0f / fmaxf(dg, 1.0f);

        const float* fRow = feat + (size_t)mrow * D + khalf;
        const float* aRow = agg  + (size_t)mrow * D + khalf;

        // LDS base for this lane's B column: n = nt*16 + nsub, addr = n*D + k + khalf
        const float* lwsBase = smem +          nsub * D + khalf;   // W_self
        const float* lwnBase = smem + D * D +  nsub * D + khalf;   // W_neigh

        // Accumulators for 8 N-tiles, initialized with fused bias.
        v8f acc[8];
#pragma unroll
        for (int nt = 0; nt < 8; ++nt) {
            const int n = nt * 16 + nsub;
            const float b = bself[n] + bneigh[n];
            acc[nt] = (v8f){b, b, b, b, b, b, b, b};
        }

        // One K-chunk: 16 ds_load_b64 into distinct regs, then interleaved
        // S/N WMMAs (per N-tile S then N, back-to-back RAW on that accumulator).
        auto do_chunk = [&](int k, v2f aS, v2f aNr) {
            const v2f aN = {aNr.x * inv, aNr.y * inv};
            v2f bS[8], bN[8];
#pragma unroll
            for (int nt = 0; nt < 8; ++nt) {
                bS[nt] = *(const v2f*)(lwsBase + nt * 16 * D + k);
                bN[nt] = *(const v2f*)(lwnBase + nt * 16 * D + k);
            }
#pragma unroll
            for (int nt = 0; nt < 8; ++nt) {
                acc[nt] = __builtin_amdgcn_wmma_f32_16x16x4_f32(
                    false, aS, false, bS[nt], (short)0, acc[nt], false, false);
                acc[nt] = __builtin_amdgcn_wmma_f32_16x16x4_f32(
                    false, aN, false, bN[nt], (short)0, acc[nt], false, false);
            }
        };

        // Software pipeline: prefetch next chunk's A while current WMMAs run.
        v2f aS_cur  = *(const v2f*)(fRow);
        v2f aNr_cur = *(const v2f*)(aRow);
#pragma unroll 2
        for (int k = 0; k < D - 4; k += 4) {
            const v2f aS_nxt  = *(const v2f*)(fRow + k + 4);
            const v2f aNr_nxt = *(const v2f*)(aRow + k + 4);
            do_chunk(k, aS_cur, aNr_cur);
            aS_cur  = aS_nxt;
            aNr_cur = aNr_nxt;
        }
        do_chunk(D - 4, aS_cur, aNr_cur);     // epilogue chunk

        // Store: per vgpr v, half-wave writes a contiguous 64B segment of
        // row M = v + 8*(lane/16) at channels nt*16 + 0..15.
        const int rowhalf = (lane >> 4) * 8;
#pragma unroll
        for (int nt = 0; nt < 8; ++nt) {
#pragma unroll
            for (int v = 0; v < 8; ++v) {
                const int m = tile * 16 + rowhalf + v;
                if (m < n_nodes)
                    out[(size_t)m * D + nt * 16 + nsub] = acc[nt][v];
            }
        }
    }
}

// ---------------------------------------------------------------------------
extern "C" void kernel_launch(void* const* d_in, const int* in_sizes, int n_in,
                              void* d_out, int out_size, void* d_ws, size_t ws_size,
                              hipStream_t stream) {
    const float* feat   = (const float*)d_in[0];
    const int*   src    = (const int*)  d_in[1];
    const int*   dst    = (const int*)  d_in[2];
    const float* Wself  = (const float*)d_in[3];
    const float* bself  = (const float*)d_in[4];
    const float* Wneigh = (const float*)d_in[5];
    const float* bneigh = (const float*)d_in[6];
    float*       out    = (float*)d_out;

    const int n_nodes = in_sizes[0] / D;
    const int n_edges = in_sizes[1];

    float* agg = (float*)d_ws;                       // n_nodes * 128 floats
    float* deg = agg + (size_t)n_nodes * D;          // n_nodes floats

    // 1) zero workspace
    {
        const int total = n_nodes * D;
        const int blocks = (total + 255) / 256;
        sage_zero_ws<<<blocks, 256, 0, stream>>>(agg, deg, n_nodes);
    }
    // 2) edge scatter (one wave per edge)
    {
        const long long threads = (long long)n_edges * 32;
        const int blocks = (int)((threads + 255) / 256);
        sage_scatter<<<blocks, 256, 0, stream>>>(feat, src, dst, agg, deg, n_edges);
    }
    // 3) fused mean + dual WMMA GEMM + bias (128KB dynamic LDS for weights)
    {
        const int n_tiles = (n_nodes + 15) / 16;
        const int blocks  = (n_tiles + 7) / 8;       // 8 waves per block
        const size_t lds_bytes = (size_t)2 * D * D * sizeof(float);  // 131072
        sage_wmma_gemm<<<blocks, 256, lds_bytes, stream>>>(
            feat, agg, deg, Wself, bself, Wneigh, bneigh, out, n_nodes, n_tiles);
    }
}